// CrossProductLayer_72189810312025
// MI455X (gfx1250) — compile-verified
//
#include <hip/hip_runtime.h>

typedef __attribute__((ext_vector_type(2))) float v2f;
typedef __attribute__((ext_vector_type(8))) float v8f;

#define NIN       128
#define NFEAT     8384      // 128 squares + 128 singles + 8128 pairs
#define PAIR_BASE 256

// One block (256 threads = 8 wave32) per batch row.
// - Stage x[b,0:128] into LDS (32 lanes x b128).
// - Squares+singles: one coalesced 1KB store (thread t -> feature t).
// - Pairs: upper-triangular 16x16 tiles of the outer product x⊗x via
//   v_wmma_f32_16x16x4_f32 (rank-1: only K=0 nonzero, 0.5 folded into A),
//   scaled by w and stored packed-triu.
//   28 off-diagonal tiles: branch-free (batched loads/stores).
//   8 diagonal tiles (one per wave): predicated on j>i.
__global__ __launch_bounds__(256)
void CrossProductLayer_kernel(const float* __restrict__ x,
                              const float* __restrict__ w,
                              float* __restrict__ out)
{
    __shared__ float xs[NIN];
    const int b   = blockIdx.x;
    const int tid = threadIdx.x;
    const float* xrow = x + (size_t)b * NIN;
    float*       orow = out + (size_t)b * NFEAT;
    const float* wp   = w + PAIR_BASE;
    float*       op   = orow + PAIR_BASE;

    // ---- stage x row into LDS: 32 lanes * float4 = 128 floats ----
    if (tid < 32) {
        ((float4*)xs)[tid] = ((const float4*)xrow)[tid];
    }
    __syncthreads();

    // ---- squares (f=0..127) and singles (f=128..255): coalesced b32 store ----
    {
        float xv  = xs[tid & 127];
        float val = (tid < 128) ? xv * xv : xv;
        orow[tid] = val * w[tid];
    }

    const int  wave = tid >> 5;
    const int  lane = tid & 31;
    const int  lh   = lane & 15;
    const bool lo   = (lane < 16);
    const int  mhi  = lo ? 0 : 8;       // row offset for lanes 16-31 in C/D

    // ======== 28 off-diagonal tiles (ti<tj): branch-free ========
    for (int t = wave; t < 28; t += 8) {
        // decode flat strict-triu tile id -> (ti, tj), ti<tj
        int ti = 0, rem = t;
        while (rem >= 7 - ti) { rem -= 7 - ti; ++ti; }
        const int tj = ti + 1 + rem;
        const int i0 = ti << 4;
        const int j0 = tj << 4;

        // Unconditional LDS reads, then mask with selects (no exec branches).
        // A (16x4): lanes 0-15 = K=0 column; lanes 16-31 = K=2 -> zero.
        // B (4x16): v[0] lanes 0-15 = row K=0; lanes 16-31 = row K=2 -> zero.
        const float av = xs[i0 + lh];
        const float bv = xs[j0 + lh];
        v2f a;  a.x  = lo ? 0.5f * av : 0.0f;  a.y  = 0.0f;
        v2f bm; bm.x = lo ? bv        : 0.0f;  bm.y = 0.0f;

        v8f c = {};
        // D[m,n] = 0.5 * x[i0+m] * x[j0+n]
        v8f d = __builtin_amdgcn_wmma_f32_16x16x4_f32(
                    false, a, false, bm, (short)0, c, false, false);

        const int ibase = i0 + mhi;
        const int j     = j0 + lh;

        int   idx[8];
        float wv[8];
        #pragma unroll
        for (int r = 0; r < 8; ++r) {
            const int i = ibase + r;
            idx[r] = ((i * (255 - i)) >> 1) + (j - i - 1);  // packed triu index
            wv[r]  = wp[idx[r]];                            // L0/L2-hit loads, batched
        }
        #pragma unroll
        for (int r = 0; r < 8; ++r) {
            op[idx[r]] = d[r] * wv[r];   // two 64B coalesced segments per store
        }
    }

    // ======== diagonal tile ti == tj == wave: predicated (j > i) ========
    {
        const int i0 = wave << 4;

        const float av = xs[i0 + lh];
        v2f a;  a.x  = lo ? 0.5f * av : 0.0f;  a.y  = 0.0f;
        v2f bm; bm.x = lo ? av        : 0.0f;  bm.y = 0.0f;

        v8f c = {};
        v8f d = __builtin_amdgcn_wmma_f32_16x16x4_f32(
                    false, a, false, bm, (short)0, c, false, false);

        const int ibase = i0 + mhi;
        const int j     = i0 + lh;
        #pragma unroll
        for (int r = 0; r < 8; ++r) {
            const int i = ibase + r;
            if (j > i) {
                const int idx = ((i * (255 - i)) >> 1) + (j - i - 1);
                op[idx] = d[r] * wp[idx];
            }
        }
    }
}

extern "C" void kernel_launch(void* const* d_in, const int* in_sizes, int n_in,
                              void* d_out, int out_size, void* d_ws, size_t ws_size,
                              hipStream_t stream) {
    const float* x = (const float*)d_in[0];
    const float* w = (const float*)d_in[1];
    float* out     = (float*)d_out;
    const int batch = in_sizes[0] / NIN;   // 16384
    CrossProductLayer_kernel<<<batch, 256, 0, stream>>>(x, w, out);
}